// OneHotEmbedding_29944511988349
// MI455X (gfx1250) — compile-verified
//
#include <hip/hip_runtime.h>

// OneHotEmbedding on MI455X (gfx1250, wave32).
//
// out[token, :] = W[:, id(token)]  (zero row when id == 0)
//
// Pure gather, 0 FLOPs. HBM floor: 103MB (W) + 64MB (out) ~= 7us @ 23.3TB/s.
// W is [512, 50257] row-major, so a token's embedding is a strided COLUMN
// (stride 201KB): naive gather costs 16.7M 4-byte reads * 128B line fill
// = ~2.1GB of L2->WGP traffic. We instead transpose W once into d_ws
// (LDS-tiled, coalesced both sides), then gather contiguous 2KB rows.
// Non-temporal hints keep the hot array (Wt) resident in the 192MB L2:
//  - phase-1 reads of W are NT (dead after transpose)
//  - output stores are NT (write-once, never re-read)

#define VOCAB 50257
#define EMBED 512

typedef __attribute__((ext_vector_type(4))) float v4f;

// ---------------------------------------------------------------------------
// Phase 1: W[512][VOCAB] -> Wt[VOCAB][512], 32x32 tiles via LDS (bank-padded).
// Block = 32x8 (8 waves), grid = (ceil(VOCAB/32), EMBED/32).
// ---------------------------------------------------------------------------
__global__ __launch_bounds__(256) void xpose_kernel(const float* __restrict__ W,
                                                    float* __restrict__ Wt) {
    __shared__ float tile[32][33];          // +1 pad: no LDS bank conflicts
    const int tx = threadIdx.x;             // 0..31
    const int ty = threadIdx.y;             // 0..7
    const int col = blockIdx.x * 32 + tx;   // vocab index (coalesced read dim)

    if (col < VOCAB) {
#pragma unroll
        for (int i = 0; i < 32; i += 8) {
            const int row = blockIdx.y * 32 + ty + i;   // d index, always < 512
            // NT load: W is dead after this pass; don't let it occupy L2.
            tile[ty + i][tx] =
                __builtin_nontemporal_load(&W[(size_t)row * VOCAB + col]);
        }
    }
    __syncthreads();

    const int d = blockIdx.y * 32 + tx;     // d index (coalesced write dim)
#pragma unroll
    for (int i = 0; i < 32; i += 8) {
        const int vr = blockIdx.x * 32 + ty + i;        // vocab row of Wt
        if (vr < VOCAB) {
            // Regular (temporal) store: Wt is the hot phase-2 working set,
            // we WANT it resident in the 192MB L2.
            Wt[(size_t)vr * EMBED + d] = tile[tx][ty + i];
        }
    }
}

// ---------------------------------------------------------------------------
// Phase 2: one wave32 per token. Row of Wt is contiguous 2KB:
// 4x global_load_b128 per lane (fully coalesced), NT b128 stores to out.
// ---------------------------------------------------------------------------
__global__ __launch_bounds__(256) void gather_fast_kernel(const int* __restrict__ tok,
                                                          const float* __restrict__ Wt,
                                                          float* __restrict__ out,
                                                          int n_tok) {
    const int w    = (int)((blockIdx.x * 256u + threadIdx.x) >> 5);  // token
    const int lane = threadIdx.x & 31;
    if (w >= n_tok) return;

    const int t = tok[w];
    v4f* __restrict__ dst = (v4f*)(out + (size_t)w * EMBED);   // 128 v4f/token

    if (t == 0) {  // PADDING_IDX -> zero row
        const v4f z = {0.f, 0.f, 0.f, 0.f};
#pragma unroll
        for (int j = 0; j < 4; ++j)
            __builtin_nontemporal_store(z, &dst[lane + 32 * j]);
        return;
    }

    const v4f* __restrict__ src = (const v4f*)(Wt + (size_t)t * EMBED);
#pragma unroll
    for (int j = 0; j < 4; ++j) {
        const v4f v = src[lane + 32 * j];                      // temporal: L2 hit
        __builtin_nontemporal_store(v, &dst[lane + 32 * j]);   // write-once
    }
}

// ---------------------------------------------------------------------------
// Fallback (only if d_ws < 103MB): direct strided-column gather out of W.
// Reads are temporal so the whole 103MB of W stays L2-resident.
// ---------------------------------------------------------------------------
__global__ __launch_bounds__(256) void gather_direct_kernel(const int* __restrict__ tok,
                                                            const float* __restrict__ W,
                                                            float* __restrict__ out,
                                                            int n_tok) {
    const int w    = (int)((blockIdx.x * 256u + threadIdx.x) >> 5);
    const int lane = threadIdx.x & 31;
    if (w >= n_tok) return;

    const int t = tok[w];
    v4f* __restrict__ dst = (v4f*)(out + (size_t)w * EMBED);

    if (t == 0) {
        const v4f z = {0.f, 0.f, 0.f, 0.f};
#pragma unroll
        for (int j = 0; j < 4; ++j)
            __builtin_nontemporal_store(z, &dst[lane + 32 * j]);
        return;
    }

    const float* __restrict__ src = W + t;   // column t, stride VOCAB floats
#pragma unroll
    for (int j = 0; j < 4; ++j) {
        const int d0 = (lane + 32 * j) * 4;
        v4f v;
        v.x = src[(size_t)(d0 + 0) * VOCAB];
        v.y = src[(size_t)(d0 + 1) * VOCAB];
        v.z = src[(size_t)(d0 + 2) * VOCAB];
        v.w = src[(size_t)(d0 + 3) * VOCAB];
        __builtin_nontemporal_store(v, &dst[lane + 32 * j]);
    }
}

extern "C" void kernel_launch(void* const* d_in, const int* in_sizes, int n_in,
                              void* d_out, int out_size, void* d_ws, size_t ws_size,
                              hipStream_t stream) {
    const int*   tok = (const int*)d_in[0];     // [8*4096] int32 token ids
    const float* W   = (const float*)d_in[1];   // [512, 50257] f32
    float*       out = (float*)d_out;           // [8*4096, 512] f32
    const int n_tok = in_sizes[0];

    const int gather_blocks = (n_tok + 7) / 8;  // 8 wave32 tokens per 256-thd block
    const size_t wt_bytes = (size_t)VOCAB * EMBED * sizeof(float);

    if (ws_size >= wt_bytes) {
        float* Wt = (float*)d_ws;
        dim3 tgrid((VOCAB + 31) / 32, EMBED / 32);
        xpose_kernel<<<tgrid, dim3(32, 8), 0, stream>>>(W, Wt);
        gather_fast_kernel<<<gather_blocks, 256, 0, stream>>>(tok, Wt, out, n_tok);
    } else {
        gather_direct_kernel<<<gather_blocks, 256, 0, stream>>>(tok, W, out, n_tok);
    }
}